// PathGCNLayer_81449759801398
// MI455X (gfx1250) — compile-verified
//
#include <hip/hip_runtime.h>
#include <hip/hip_bf16.h>

typedef __attribute__((ext_vector_type(2))) float v2f;
typedef __attribute__((ext_vector_type(4))) float v4f;
typedef __attribute__((ext_vector_type(8))) float v8f;

#define D_DIM 128
#define P_NUM 6
#define L_LEN 6

// ---------------------------------------------------------------------------
// Kernel 1: acc[n,d] = (1/6) * sum_{p,l} feats[paths[p,n,l], d] * w[l,d]
// One wave (32 lanes) per node; lane owns float4 slice of D=128.
// feats (25.6 MB) is L2-resident, so the 922 MB of gathered reads hit L2.
// ---------------------------------------------------------------------------
__global__ __launch_bounds__(256) void pathgcn_gather_kernel(
    const float* __restrict__ feats,   // (N, 128)
    const int*   __restrict__ paths,   // (P, N, L) int32
    const float* __restrict__ w,       // (L, 128)
    float*       __restrict__ results, // (N, 128) workspace
    int N)
{
    const int lane = threadIdx.x & 31;
    const int wave = threadIdx.x >> 5;
    // force node to SGPR so index loads become scalar and feats loads use saddr
    const int node = __builtin_amdgcn_readfirstlane((int)(blockIdx.x * 8 + wave));
    if (node >= N) return;

    // preload path weights for this lane's 4 columns
    v4f wv[L_LEN];
#pragma unroll
    for (int l = 0; l < L_LEN; ++l)
        wv[l] = *(const v4f*)(w + l * D_DIM + lane * 4);

    v4f acc = {0.f, 0.f, 0.f, 0.f};
#pragma unroll
    for (int p = 0; p < P_NUM; ++p) {
        const int* idxp = paths + ((size_t)p * (size_t)N + (size_t)node) * L_LEN;
#pragma unroll
        for (int l = 0; l < L_LEN; ++l) {
            const int idx = __builtin_amdgcn_readfirstlane(idxp[l]);
            const v4f f = *(const v4f*)(feats + (size_t)idx * D_DIM + lane * 4);
            acc += f * wv[l];
        }
    }
    const float inv = 1.0f / (float)P_NUM;
    acc *= inv;
    *(v4f*)(results + (size_t)node * D_DIM + lane * 4) = acc;
}

// ---------------------------------------------------------------------------
// Kernel 2: out = relu(results @ fcw^T) via V_WMMA_F32_16X16X4_F32.
// Each wave: 16 nodes x 64 out-features (4 accumulator tiles), K loop by 4.
// A layout (16x4 f32): lanes 0-15 hold M=row, K={k,k+1}; lanes 16-31 K={k+2,k+3}.
// B layout mirrors A with N=row (B[k][n] = fcw[n][k], fcw is (out,in) row-major).
// C/D layout: VGPR v -> M = v + 8*(lane>=16), Ncol = lane&15.
// ---------------------------------------------------------------------------
__global__ __launch_bounds__(256) void pathgcn_gemm_relu_kernel(
    const float* __restrict__ results, // (N, 128)
    const float* __restrict__ fcw,     // (128, 128) row-major (out, in)
    float*       __restrict__ out,     // (N, 128)
    int N)
{
    const int lane = threadIdx.x & 31;
    const int wave = threadIdx.x >> 5;
    const int mtiles = (N + 15) >> 4;
    const int mt = (int)blockIdx.x * 4 + (wave >> 1);
    if (mt >= mtiles) return;            // wave-uniform exit; EXEC stays all-ones

    const int m0 = mt << 4;
    const int j0 = (wave & 1) * 64;      // this wave's 4 tiles of 16 out-features

    const int row = lane & 15;
    const int kh  = lane >> 4;           // K sub-offset: 0 -> {k,k+1}, 1 -> {k+2,k+3}

    int arow = m0 + row;                 // clamp tail rows (loads must stay in-bounds,
    if (arow >= N) arow = N - 1;         // EXEC must remain all-ones for WMMA)
    const float* Ap = results + (size_t)arow * D_DIM + kh * 2;
    const float* Bp = fcw + (size_t)(j0 + row) * D_DIM + kh * 2;

    v8f c0 = {}, c1 = {}, c2 = {}, c3 = {};
#pragma unroll 4
    for (int k = 0; k < D_DIM; k += 4) {
        const v2f a  = *(const v2f*)(Ap + k);
        const v2f b0 = *(const v2f*)(Bp + k);
        const v2f b1 = *(const v2f*)(Bp + 16 * D_DIM + k);
        const v2f b2 = *(const v2f*)(Bp + 32 * D_DIM + k);
        const v2f b3 = *(const v2f*)(Bp + 48 * D_DIM + k);
        c0 = __builtin_amdgcn_wmma_f32_16x16x4_f32(false, a, false, b0, (short)0, c0, false, false);
        c1 = __builtin_amdgcn_wmma_f32_16x16x4_f32(false, a, false, b1, (short)0, c1, false, false);
        c2 = __builtin_amdgcn_wmma_f32_16x16x4_f32(false, a, false, b2, (short)0, c2, false, false);
        c3 = __builtin_amdgcn_wmma_f32_16x16x4_f32(false, a, false, b3, (short)0, c3, false, false);
    }

    const int ncol = row;
    const int mofs = kh * 8;
    const v8f cs[4] = {c0, c1, c2, c3};
#pragma unroll
    for (int jt = 0; jt < 4; ++jt) {
#pragma unroll
        for (int v = 0; v < 8; ++v) {
            const int m = m0 + v + mofs;
            if (m < N) {
                float x = cs[jt][v];
                out[(size_t)m * D_DIM + (j0 + jt * 16 + ncol)] = x > 0.f ? x : 0.f;
            }
        }
    }
}

// ---------------------------------------------------------------------------
// Host launcher. Input order (reference setup_inputs):
//   0: feats (N*128 f32)   1: paths (6*N*6 int32, JAX x64 disabled)
//   2: init_feats (unused) 3: flag (==1, the only value that type-checks)
//   4: path_weight1 (6*128)  5: path_weight2 (unused)  6: fc_weight (128*128)
// ---------------------------------------------------------------------------
extern "C" void kernel_launch(void* const* d_in, const int* in_sizes, int n_in,
                              void* d_out, int out_size, void* d_ws, size_t ws_size,
                              hipStream_t stream) {
    const float* feats = (const float*)d_in[0];
    const int*   paths = (const int*)d_in[1];
    const float* pw1   = (const float*)d_in[4];
    const float* fcw   = (const float*)d_in[6];
    float*       out   = (float*)d_out;
    float*       results = (float*)d_ws;   // N*128 f32 scratch

    const int N = in_sizes[0] / D_DIM;

    dim3 blk(256);
    dim3 g1((N + 7) / 8);                 // 8 waves/block, 1 node/wave
    pathgcn_gather_kernel<<<g1, blk, 0, stream>>>(feats, paths, pw1, results, N);

    const int mtiles = (N + 15) / 16;
    dim3 g2((mtiles + 3) / 4);            // 8 waves/block: 4 m-tiles x 2 j-halves
    pathgcn_gemm_relu_kernel<<<g2, blk, 0, stream>>>(results, fcw, out, N);
}